// RobustTripletLoss_final_61564061221311
// MI455X (gfx1250) — compile-verified
//
#include <hip/hip_runtime.h>
#include <hip/hip_bf16.h>
#include <cstdint>
#include <cmath>

#define NROWS 4096
#define DDIM  2048
#define NCLS  128
#define MARGIN_F 0.3f

typedef __attribute__((ext_vector_type(2))) float v2f;
typedef __attribute__((ext_vector_type(8))) float v8f;

// Typed pointers for the gfx1250 async LDS-load builtin:
//   void __builtin_amdgcn_global_load_async_to_lds_b128(v4i AS1*, v4i AS3*, imm, imm)
typedef int rtl_v4i __attribute__((vector_size(16)));
typedef rtl_v4i __attribute__((address_space(1))) as1_v4i;
typedef rtl_v4i __attribute__((address_space(3))) as3_v4i;

#if __has_builtin(__builtin_amdgcn_global_load_async_to_lds_b128)
#define RTL_ASYNC_LDS 1
#endif

#if __has_builtin(__builtin_amdgcn_s_wait_asynccnt)
#define RTL_ASYNC_WAIT(n) __builtin_amdgcn_s_wait_asynccnt(n)
#else
#define RTL_ASYNC_WAIT(n) asm volatile("s_wait_asynccnt %0" ::"n"(n))
#endif

// ---------------------------------------------------------------------------
// Row squared norms: sq[i] = sum_k X[i][k]^2
// ---------------------------------------------------------------------------
__global__ __launch_bounds__(256) void rtl_sqnorm_kernel(const float* __restrict__ X,
                                                         float* __restrict__ sq) {
    __shared__ float red[256];
    const int i = blockIdx.x;
    const int tid = threadIdx.x;
    float s = 0.0f;
#pragma unroll
    for (int t = 0; t < DDIM / 256; ++t) {
        float v = X[(size_t)i * DDIM + tid + t * 256];
        s += v * v;
    }
    red[tid] = s;
    __syncthreads();
    for (int off = 128; off > 0; off >>= 1) {
        if (tid < off) red[tid] += red[tid + off];
        __syncthreads();
    }
    if (tid == 0) sq[i] = red[0];
}

// ---------------------------------------------------------------------------
// pred_cls[r] = argmax_c prediction[r][c]  (first max wins, like jnp.argmax)
// ---------------------------------------------------------------------------
__global__ void rtl_predcls_kernel(const float* __restrict__ P, int* __restrict__ pc) {
    int r = blockIdx.x * blockDim.x + threadIdx.x;
    if (r >= NROWS) return;
    const float* row = P + (size_t)r * NCLS;
    float best = row[0];
    int bi = 0;
    for (int c = 1; c < NCLS; ++c) {
        float v = row[c];
        if (v > best) { best = v; bi = c; }
    }
    pc[r] = bi;
}

// ---------------------------------------------------------------------------
// Gram + distance: dist[r][c] = sqrt(max(sq[r]+sq[c]-2*(X_r . X_c), 1e-12))
// 8 waves/block; wave w -> 16 rows, block -> 64 cols. fp32 WMMA 16x16x4.
// Double-buffered LDS staging via GLOBAL_LOAD_ASYNC_TO_LDS_B128 when available.
// ---------------------------------------------------------------------------
#define KC  32
#define LDK 36   // padded LDS row stride (floats); 144B keeps rows 16B-aligned
#define NCHUNK (DDIM / KC)

__device__ __forceinline__ void rtl_stage_chunk(const float* __restrict__ X,
                                                int rowBase, int colBase, int k0,
                                                float* __restrict__ tRb,
                                                float* __restrict__ tCb, int tid) {
#pragma unroll
    for (int it = 0; it < 6; ++it) {            // 192 rows * 8 quads / 256 thr
        int q  = tid + it * 256;
        int r  = q >> 3;
        int qc = q & 7;
        int grow = (r < 128) ? (rowBase + r) : (colBase + (r - 128));
        const float* gp = X + (size_t)grow * DDIM + k0 + qc * 4;
        float* lp = (r < 128) ? (tRb + r * LDK + qc * 4)
                              : (tCb + (r - 128) * LDK + qc * 4);
#ifdef RTL_ASYNC_LDS
        __builtin_amdgcn_global_load_async_to_lds_b128(
            (as1_v4i*)(unsigned long long)gp,
            (as3_v4i*)(unsigned)(unsigned long long)lp, 0, 0);
#else
        const float4 v = *(const float4*)gp;
        *(float4*)lp = v;
#endif
    }
}

__global__ __launch_bounds__(256) void rtl_gram_dist_kernel(const float* __restrict__ X,
                                                            const float* __restrict__ sq,
                                                            float* __restrict__ dist) {
    __shared__ float tR[2][128 * LDK];  // 8 row tiles of 16 rows, x2 buffers
    __shared__ float tC[2][64 * LDK];   // shared 64-column panel, x2 buffers

    const int tid  = threadIdx.x;
    const int lane = tid & 31;
    const int w    = tid >> 5;           // wave id 0..7
    const int m    = lane & 15;          // row/col within 16
    const int koff = (lane >> 4) << 1;   // K sub-offset 0 or 2 (A/B fp32 layout)
    const int rowBase = blockIdx.y * 128;
    const int colBase = blockIdx.x * 64;

    v8f acc[4];
#pragma unroll
    for (int ct = 0; ct < 4; ++ct) acc[ct] = 0.0f;

    // Prime buffer 0 with chunk 0.
    rtl_stage_chunk(X, rowBase, colBase, 0, tR[0], tC[0], tid);

    for (int c = 0; c < NCHUNK; ++c) {
        const int buf = c & 1;
        // Stream next chunk into the other buffer while we compute this one.
        if (c + 1 < NCHUNK) {
            rtl_stage_chunk(X, rowBase, colBase, (c + 1) * KC,
                            tR[buf ^ 1], tC[buf ^ 1], tid);
            RTL_ASYNC_WAIT(6);   // older batch (this chunk) has landed in LDS
        } else {
            RTL_ASYNC_WAIT(0);   // drain
        }
        __syncthreads();          // all waves' data for chunk c visible

        const float* tRb = tR[buf];
        const float* tCb = tC[buf];
#pragma unroll
        for (int kk = 0; kk < KC; kk += 4) {
            // A 16x4 fp32: lane m holds A[m][koff], A[m][koff+1]
            v2f a = *(const v2f*)(tRb + (w * 16 + m) * LDK + kk + koff);
#pragma unroll
            for (int ct = 0; ct < 4; ++ct) {
                // B 4x16 fp32: lane m holds B[koff][m], B[koff+1][m]
                v2f b = *(const v2f*)(tCb + (ct * 16 + m) * LDK + kk + koff);
                acc[ct] = __builtin_amdgcn_wmma_f32_16x16x4_f32(
                    false, a, false, b, (short)0, acc[ct], false, false);
            }
        }
        __syncthreads();          // safe to overwrite buf next iteration
    }

    // Epilogue: fuse sqrt(clip(sq_r + sq_c - 2G)) into the store.
    const int rhalf = (lane >> 4) << 3;   // C layout: lanes16-31 hold M+8
    float sqr[8];
#pragma unroll
    for (int v = 0; v < 8; ++v) sqr[v] = sq[rowBase + w * 16 + rhalf + v];
#pragma unroll
    for (int ct = 0; ct < 4; ++ct) {
        const int cc = colBase + ct * 16 + m;
        const float sqc = sq[cc];
#pragma unroll
        for (int v = 0; v < 8; ++v) {
            const int r = rowBase + w * 16 + rhalf + v;
            float d2 = sqr[v] + sqc - 2.0f * acc[ct][v];
            dist[(size_t)r * NROWS + cc] = sqrtf(fmaxf(d2, 1e-12f));
        }
    }
}

// ---------------------------------------------------------------------------
// JAX threefry2x32 (20 rounds) — matches jax.random key semantics.
// ---------------------------------------------------------------------------
__device__ __forceinline__ unsigned rtl_rotl32(unsigned x, int n) {
    return (x << n) | (x >> (32 - n));
}

__device__ __forceinline__ void rtl_threefry2x32(unsigned k0, unsigned k1,
                                                 unsigned c0, unsigned c1,
                                                 unsigned& o0, unsigned& o1) {
    const unsigned ks2 = k0 ^ k1 ^ 0x1BD11BDAu;
    unsigned x0 = c0 + k0, x1 = c1 + k1;
#define TFR(r) { x0 += x1; x1 = rtl_rotl32(x1, r); x1 ^= x0; }
    TFR(13) TFR(15) TFR(26) TFR(6)
    x0 += k1;  x1 += ks2 + 1u;
    TFR(17) TFR(29) TFR(16) TFR(24)
    x0 += ks2; x1 += k0 + 2u;
    TFR(13) TFR(15) TFR(26) TFR(6)
    x0 += k0;  x1 += k1 + 3u;
    TFR(17) TFR(29) TFR(16) TFR(24)
    x0 += k1;  x1 += ks2 + 4u;
    TFR(13) TFR(15) TFR(26) TFR(6)
    x0 += ks2; x1 += k0 + 5u;
#undef TFR
    o0 = x0; o1 = x1;
}

__device__ __forceinline__ float rtl_bits_to_gumbel(unsigned b) {
    // jax uniform(minval=tiny, maxval=1): [1,2) mantissa trick, then affine
    float f = __uint_as_float((b >> 9) | 0x3f800000u) - 1.0f;
    const float tiny = 1.17549435e-38f;
    float u = f * (1.0f - tiny) + tiny;
    return -logf(-logf(u));
}

// Block-wide gumbel-argmax over j=0..4095 (mask = pos bits; optionally j!=i).
// Bits layout follows jax: pair p -> (bits[p], bits[2048+p]).
__device__ int rtl_gumbel_scan(unsigned k0, unsigned k1, unsigned pmask,
                               int tid, int i, bool exclude_self,
                               float* rv, int* ri) {
    float bv = -INFINITY;
    int bi = 0x7FFFFFFF;
#pragma unroll
    for (int t = 0; t < 8; ++t) {
        int p = tid + t * 256;
        unsigned b0, b1;
        rtl_threefry2x32(k0, k1, (unsigned)p, (unsigned)(2048 + p), b0, b1);
        int j0 = p, j1 = 2048 + p;           // slots t and t+8 in pmask
        bool m0 = ((pmask >> t) & 1u) && (!exclude_self || j0 != i);
        bool m1 = ((pmask >> (t + 8)) & 1u) && (!exclude_self || j1 != i);
        float g0 = m0 ? rtl_bits_to_gumbel(b0) : -INFINITY;
        float g1 = m1 ? rtl_bits_to_gumbel(b1) : -INFINITY;
        if (g0 > bv || (g0 == bv && j0 < bi)) { bv = g0; bi = j0; }
        if (g1 > bv || (g1 == bv && j1 < bi)) { bv = g1; bi = j1; }
    }
    rv[tid] = bv; ri[tid] = bi;
    __syncthreads();
    for (int off = 128; off > 0; off >>= 1) {
        if (tid < off) {
            float ov = rv[tid + off]; int oi = ri[tid + off];
            if (ov > rv[tid] || (ov == rv[tid] && oi < ri[tid])) {
                rv[tid] = ov; ri[tid] = oi;
            }
        }
        __syncthreads();
    }
    int win = ri[0];
    __syncthreads();
    return win;
}

// ---------------------------------------------------------------------------
// Per-row: top-12 negatives, RNG choices, triplet branch logic.
// One 256-thread block per row.
// ---------------------------------------------------------------------------
__global__ __launch_bounds__(256) void rtl_per_row_kernel(
    const float* __restrict__ dist, const int* __restrict__ targets,
    const float* __restrict__ prob, const int* __restrict__ predcls,
    const float* __restrict__ thrp, float* __restrict__ aps,
    float* __restrict__ ansv, float* __restrict__ lossv) {
    __shared__ float rv[256];
    __shared__ int   ri[256];
    __shared__ int   s_picked[12];
    __shared__ unsigned s_keys[14];  // [0..1]=kp, [2+2d,3+2d]=draw key d

    const int tid = threadIdx.x;
    const int i   = blockIdx.x;
    const float th = *thrp;
    const int ti   = targets[i];
    const size_t rowOff = (size_t)i * NROWS;

    float ndv[16];
    unsigned pmask = 0;
#pragma unroll
    for (int s = 0; s < 16; ++s) {
        int j = tid + s * 256;
        float dj = dist[rowOff + j];
        bool pos = (targets[j] == ti);   // diagonal is pos -> excluded from negs
        if (pos) pmask |= (1u << s);
        ndv[s] = pos ? INFINITY : dj;
    }

    // neg_rank[i, 0..11]: 12 smallest negative distances (ties -> lower index)
    for (int it = 0; it < 12; ++it) {
        float bv = INFINITY;
        int bi = 0x7FFFFFFF;
#pragma unroll
        for (int s = 0; s < 16; ++s) {
            int j = tid + s * 256;
            float v = ndv[s];
            if (v < bv || (v == bv && j < bi)) { bv = v; bi = j; }
        }
        rv[tid] = bv; ri[tid] = bi;
        __syncthreads();
        for (int off = 128; off > 0; off >>= 1) {
            if (tid < off) {
                float ov = rv[tid + off]; int oi = ri[tid + off];
                if (ov < rv[tid] || (ov == rv[tid] && oi < ri[tid])) {
                    rv[tid] = ov; ri[tid] = oi;
                }
            }
            __syncthreads();
        }
        int win = ri[0];
        if (tid == 0) s_picked[it] = win;
        if ((win & 255) == tid) {
            int slot = win >> 8;
#pragma unroll
            for (int s = 0; s < 16; ++s)
                if (s == slot) ndv[s] = INFINITY;
        }
        __syncthreads();
    }

    // RNG key derivation: row_keys = split(key(42), 4096); kp,kr = split(rk);
    // draw keys = split(kr, 6).  (threefry key for jax.random.key(42) = (0,42))
    if (tid == 0) {
        unsigned rk0, rk1, scratch;
        if (i < 2048) {
            rtl_threefry2x32(0u, 42u, (unsigned)(2 * i),     (unsigned)(4096 + 2 * i),     rk0, scratch);
            rtl_threefry2x32(0u, 42u, (unsigned)(2 * i + 1), (unsigned)(4096 + 2 * i + 1), rk1, scratch);
        } else {
            int p = 2 * i - 4096;
            rtl_threefry2x32(0u, 42u, (unsigned)p,       (unsigned)(4096 + p),     scratch, rk0);
            rtl_threefry2x32(0u, 42u, (unsigned)(p + 1), (unsigned)(4096 + p + 1), scratch, rk1);
        }
        unsigned y00, y10, y01, y11;
        rtl_threefry2x32(rk0, rk1, 0u, 2u, y00, y10);
        rtl_threefry2x32(rk0, rk1, 1u, 3u, y01, y11);
        s_keys[0] = y00; s_keys[1] = y01;          // kp
        unsigned kr0 = y10, kr1 = y11;             // kr
        unsigned g0[6], g1[6];
#pragma unroll
        for (int d = 0; d < 6; ++d)
            rtl_threefry2x32(kr0, kr1, (unsigned)d, (unsigned)(6 + d), g0[d], g1[d]);
        s_keys[2]  = g0[0]; s_keys[3]  = g0[1];    // dk0
        s_keys[4]  = g0[2]; s_keys[5]  = g0[3];    // dk1
        s_keys[6]  = g0[4]; s_keys[7]  = g0[5];    // dk2
        s_keys[8]  = g1[0]; s_keys[9]  = g1[1];    // dk3
        s_keys[10] = g1[2]; s_keys[11] = g1[3];    // dk4
        s_keys[12] = g1[4]; s_keys[13] = g1[5];    // dk5
    }
    __syncthreads();

    // rp = _choice(kp, pos & idx!=i); 6 draws with mask = pos
    int rp = rtl_gumbel_scan(s_keys[0], s_keys[1], pmask, tid, i, true, rv, ri);
    int rd[6];
#pragma unroll
    for (int d = 0; d < 6; ++d)
        rd[d] = rtl_gumbel_scan(s_keys[2 + 2 * d], s_keys[3 + 2 * d],
                                pmask, tid, i, false, rv, ri);

    if (tid == 0) {
        int hn    = s_picked[0];
        float ap0 = dist[rowOff + rp];
        float an0 = dist[rowOff + hn];
        bool p_pos = prob[rp] >= th;
        bool p_neg = prob[hn] >= th;
        bool is_FN = (predcls[hn] == ti);

        int sel = 10; bool found = false;
        for (int k = 0; k < 11; ++k) {
            if (!found && prob[s_picked[1 + k]] >= th) { sel = k; found = true; }
        }
        int hn_new = s_picked[1 + sel];
        float apB = (ap0 + an0) * 0.5f;
        float anB = dist[rowOff + hn_new];

        int selc = 5; bool foundc = false;
        for (int d = 0; d < 6; ++d) {
            bool valid = (rd[d] != i) && (prob[rd[d]] >= th);
            if (!foundc && valid) { selc = d; foundc = true; }
        }
        int rp_new = rd[selc];
        float apC = dist[rowOff + rp_new];
        float anC = (ap0 + an0) * 0.5f;

        bool useB    = p_pos && !p_neg && is_FN;
        bool useC    = (!p_pos && p_neg) || (!p_pos && !p_neg && !is_FN);
        bool inverse = !p_pos && !p_neg && is_FN;

        float ap = useB ? apB : (useC ? apC : ap0);
        float an = useB ? anB : (useC ? anC : an0);
        float pl = inverse ? fmaxf(an - ap + MARGIN_F, 0.0f)
                           : fmaxf(ap - an + MARGIN_F, 0.0f);
        aps[i] = ap; ansv[i] = an; lossv[i] = pl;
    }
}

// ---------------------------------------------------------------------------
// Final reduction -> (loss, correct, cnt) as 3 floats.
// ---------------------------------------------------------------------------
__global__ __launch_bounds__(256) void rtl_finalize_kernel(
    const float* __restrict__ prob, const float* __restrict__ thrp,
    const float* __restrict__ aps, const float* __restrict__ ansv,
    const float* __restrict__ lossv, float* __restrict__ out) {
    __shared__ float sl[256];
    __shared__ int   sc[256];
    __shared__ int   sr[256];
    const int tid = threadIdx.x;
    const float th = *thrp;
    float ls = 0.0f; int cnt = 0, corr = 0;
    for (int r = tid; r < NROWS; r += 256) {
        bool conf = prob[r] >= th;
        if (conf) {
            cnt++;
            ls += lossv[r];
            if (ansv[r] >= aps[r]) corr++;
        }
    }
    sl[tid] = ls; sc[tid] = cnt; sr[tid] = corr;
    __syncthreads();
    for (int off = 128; off > 0; off >>= 1) {
        if (tid < off) {
            sl[tid] += sl[tid + off];
            sc[tid] += sc[tid + off];
            sr[tid] += sr[tid + off];
        }
        __syncthreads();
    }
    if (tid == 0) {
        int c = sc[0];
        float loss = (c > 0) ? (sl[0] / (float)(c > 1 ? c : 1)) : 0.0f;
        out[0] = loss;
        out[1] = (float)sr[0];
        out[2] = (float)c;
    }
}

// ---------------------------------------------------------------------------
extern "C" void kernel_launch(void* const* d_in, const int* in_sizes, int n_in,
                              void* d_out, int out_size, void* d_ws, size_t ws_size,
                              hipStream_t stream) {
    (void)in_sizes; (void)n_in; (void)out_size; (void)ws_size;
    const float* inputs     = (const float*)d_in[0];
    const float* prediction = (const float*)d_in[1];
    const int*   targets    = (const int*)d_in[2];
    // d_in[3] = true_targets: unused by the reference computation
    const float* prob       = (const float*)d_in[4];
    const float* thr        = (const float*)d_in[5];
    float* out = (float*)d_out;

    char* ws = (char*)d_ws;
    const size_t distBytes = (size_t)NROWS * NROWS * sizeof(float);
    float* dist  = (float*)ws;
    float* sq    = (float*)(ws + distBytes);
    int*   pc    = (int*)  (ws + distBytes + 1 * NROWS * 4);
    float* aps   = (float*)(ws + distBytes + 2 * NROWS * 4);
    float* ansv  = (float*)(ws + distBytes + 3 * NROWS * 4);
    float* lossv = (float*)(ws + distBytes + 4 * NROWS * 4);

    rtl_sqnorm_kernel<<<NROWS, 256, 0, stream>>>(inputs, sq);
    rtl_predcls_kernel<<<NROWS / 256, 256, 0, stream>>>(prediction, pc);
    rtl_gram_dist_kernel<<<dim3(NROWS / 64, NROWS / 128), 256, 0, stream>>>(inputs, sq, dist);
    rtl_per_row_kernel<<<NROWS, 256, 0, stream>>>(dist, targets, prob, pc, thr,
                                                  aps, ansv, lossv);
    rtl_finalize_kernel<<<1, 256, 0, stream>>>(prob, thr, aps, ansv, lossv, out);
}